// GIN_1layer_71949292143001
// MI455X (gfx1250) — compile-verified
//
#include <hip/hip_runtime.h>

typedef __attribute__((ext_vector_type(2))) float v2f;
typedef __attribute__((ext_vector_type(8))) float v8f;

#define N_NODES   100000
#define D_FEAT    128
#define N_EDGES   1600000
#define N_CLASSES 32
#define M_TILES   (N_NODES / 16)   // 6250 full 16-row tiles (exact)

// ---------------------------------------------------------------------------
// Kernel 1: y = x @ W^T (f32 WMMA, exact), out = y + bias
// One wave per 16-row tile; two 16-wide N tiles cover all 32 classes.
// A-matrix 16x4 f32: lanes 0-15 rows, VGPR0/1 hold K={0,1} (lo lanes) /
// K={2,3} (hi lanes). B mirrors with N across lanes. (ISA 7.12.2)
// ---------------------------------------------------------------------------
__global__ __launch_bounds__(128)
void gin_proj_wmma(const float* __restrict__ x, const float* __restrict__ W,
                   const float* __restrict__ bias, float* __restrict__ y,
                   float* __restrict__ out) {
    const int lane  = threadIdx.x & 31;
    const int wave  = threadIdx.x >> 5;
    const int mtile = blockIdx.x * 4 + wave;
    if (mtile >= M_TILES) return;            // wave-uniform: EXEC stays all-1s

    const int col   = lane & 15;             // A-row index AND B/C column index
    const int khalf = (lane >> 4) << 1;      // lanes 16-31 supply K+2,K+3

    const float* xrow  = x + (mtile * 16 + col) * D_FEAT;
    const float* wrow0 = W + col * D_FEAT;          // classes 0..15
    const float* wrow1 = W + (col + 16) * D_FEAT;   // classes 16..31

    v8f acc0 = {};
    v8f acc1 = {};
#pragma unroll 4
    for (int k0 = 0; k0 < D_FEAT; k0 += 4) {
        v2f a  = *(const v2f*)(xrow  + k0 + khalf);
        v2f b0 = *(const v2f*)(wrow0 + k0 + khalf);
        v2f b1 = *(const v2f*)(wrow1 + k0 + khalf);
        // (neg_a, A, neg_b, B, c_mod, C, reuse_a, reuse_b)
        acc0 = __builtin_amdgcn_wmma_f32_16x16x4_f32(false, a, false, b0,
                                                     (short)0, acc0, false, false);
        acc1 = __builtin_amdgcn_wmma_f32_16x16x4_f32(false, a, false, b1,
                                                     (short)0, acc1, false, false);
    }

    const float bias0 = bias[col];
    const float bias1 = bias[col + 16];
    // C/D layout: VGPR r -> row r (lanes 0-15) / row 8+r (lanes 16-31)
    const int rbase = mtile * 16 + ((lane >> 4) << 3);
#pragma unroll
    for (int r = 0; r < 8; ++r) {
        const int row = rbase + r;
        y[row * N_CLASSES + col]        = acc0[r];
        y[row * N_CLASSES + col + 16]   = acc1[r];
        out[row * N_CLASSES + col]      = acc0[r] + bias0;
        out[row * N_CLASSES + col + 16] = acc1[r] + bias1;
    }
}

// ---------------------------------------------------------------------------
// Kernel 2: out[dst[e]] += y[src[e]]  — lane = class (wave32 == 32 classes).
// y (12.8 MB) and out (12.8 MB) are L2-resident; atomics resolve in L2.
// ---------------------------------------------------------------------------
__global__ __launch_bounds__(256)
void gin_edge_scatter(const int* __restrict__ src, const int* __restrict__ dst,
                      const float* __restrict__ y, float* __restrict__ out) {
    const int  c       = threadIdx.x & 31;
    long long  t       = (long long)blockIdx.x * blockDim.x + threadIdx.x;
    const long long nw = ((long long)gridDim.x * blockDim.x) >> 5;
    for (long long e = t >> 5; e < N_EDGES; e += nw) {
        const int   s = src[e];                 // wave-uniform address, L2 hit
        const int   d = dst[e];
        const float v = y[(long long)s * N_CLASSES + c];   // coalesced 128B
        atomicAdd(out + (long long)d * N_CLASSES + c, v);  // global_atomic_add_f32
    }
}

extern "C" void kernel_launch(void* const* d_in, const int* in_sizes, int n_in,
                              void* d_out, int out_size, void* d_ws, size_t ws_size,
                              hipStream_t stream) {
    const float* x          = (const float*)d_in[0];
    const int*   edge_index = (const int*)  d_in[1];   // [2, N_EDGES]
    const float* W          = (const float*)d_in[2];   // [32, 128] row-major
    const float* b          = (const float*)d_in[3];   // [32]
    float*       out        = (float*)d_out;           // [N_NODES, 32]
    float*       y          = (float*)d_ws;            // [N_NODES, 32] scratch

    const int* src = edge_index;                       // row 0
    const int* dst = edge_index + N_EDGES;             // row 1

    const int proj_blocks = (M_TILES + 3) / 4;         // 4 waves (16-row tiles)/block
    gin_proj_wmma<<<proj_blocks, 128, 0, stream>>>(x, W, b, y, out);

    const int edge_blocks = (N_EDGES * 32) / 256;      // one wave per edge pass
    gin_edge_scatter<<<edge_blocks, 256, 0, stream>>>(src, dst, y, out);
}